// GMGSRegressionLoss5_32289564131835
// MI455X (gfx1250) — compile-verified
//
#include <hip/hip_runtime.h>

// ---------------------------------------------------------------------------
// GMGS regression loss, MI455X (gfx1250, wave32).
// Memory-bound streaming reduction: 201 MB @ 23.3 TB/s ~ 8.6 us floor.
// Pass 1: grid-stride b128 non-temporal loads, per-thread f32 partials,
//         wave reduction via V_WMMA_F32_16X16X4_F32 (ones-matrix trick),
//         LDS cross-wave reduce, one partial per block -> d_ws.
// Pass 2: single block reduces block partials (double accum), applies the
//         common softmax-normalization factor and 1/B. Deterministic.
// ---------------------------------------------------------------------------

typedef float f4  __attribute__((ext_vector_type(4)));
typedef float v2f __attribute__((ext_vector_type(2)));
typedef float v8f __attribute__((ext_vector_type(8)));

#define THREADS 256
#define WAVES   (THREADS / 32)
#define MAX_BLOCKS 2048

// class in {0..3}: clip(floor(x), -1, 2) + 1
__device__ __forceinline__ int cls4(float x) {
    float f = floorf(x);
    f = fminf(fmaxf(f, -1.0f), 2.0f);
    return (int)f + 1;
}

__global__ void gmgs_partial_kernel(const float* __restrict__ pred,
                                    const float* __restrict__ tru,
                                    const float* __restrict__ score,
                                    float* __restrict__ partials,
                                    long long n) {
    __shared__ float s_w[16];     // unnormalized exp(-score) table
    __shared__ float s_wave[WAVES];

    if (threadIdx.x < 16) s_w[threadIdx.x] = expf(-score[threadIdx.x]);
    __syncthreads();

    float part = 0.0f;

    const long long n4     = n >> 2;
    const long long stride = (long long)gridDim.x * blockDim.x;
    const f4* __restrict__ p4 = (const f4*)pred;
    const f4* __restrict__ t4 = (const f4*)tru;

    for (long long i = (long long)blockIdx.x * blockDim.x + threadIdx.x;
         i < n4; i += stride) {
        f4 p = __builtin_nontemporal_load(p4 + i);   // global_load_b128, TH=NT
        f4 t = __builtin_nontemporal_load(t4 + i);
#pragma unroll
        for (int k = 0; k < 4; ++k) {
            float pv = p[k], tv = t[k];
            float d  = pv - tv;
            int   idx = cls4(tv) * 4 + cls4(pv);     // sm[true_cls, pred_cls]
            part += d * d * s_w[idx];
        }
    }

    // scalar tail (n % 4 elements), block 0 only; re-converges before WMMA
    long long tail = n & 3;
    if (blockIdx.x == 0 && (long long)threadIdx.x < tail) {
        long long i = (n4 << 2) + threadIdx.x;
        float pv = pred[i], tv = tru[i];
        float d  = pv - tv;
        part += d * d * s_w[cls4(tv) * 4 + cls4(pv)];
    }

    // ---- wave32 reduction via V_WMMA_F32_16X16X4_F32 -----------------------
    // A (16x4 f32, 2 VGPRs): VGPR0 = per-lane partial, VGPR1 = 0.
    //   => A[m][0] = partial(lane m), A[m][2] = partial(lane m+16), rest 0.
    // B (4x16) = all ones  =>  D[m][n] = partial(m) + partial(m+16), all n.
    // Lanes 0-15 hold rows 0-7 in c[0..7]; lanes 16-31 hold rows 8-15.
    // waveSum = sum(c) + shfl_xor_16(sum(c)).  EXEC is all ones here.
    v2f a; a[0] = part; a[1] = 0.0f;
    v2f b; b[0] = 1.0f; b[1] = 1.0f;
    v8f c = {};
    c = __builtin_amdgcn_wmma_f32_16x16x4_f32(
            /*neg_a=*/false, a, /*neg_b=*/false, b,
            /*c_mod=*/(short)0, c, /*reuse_a=*/false, /*reuse_b=*/false);
    float s8 = c[0] + c[1] + c[2] + c[3] + c[4] + c[5] + c[6] + c[7];
    float waveSum = s8 + __shfl_xor(s8, 16, 32);

    if ((threadIdx.x & 31) == 0) s_wave[threadIdx.x >> 5] = waveSum;
    __syncthreads();
    if (threadIdx.x == 0) {
        float bsum = 0.0f;
#pragma unroll
        for (int w = 0; w < WAVES; ++w) bsum += s_wave[w];
        partials[blockIdx.x] = bsum;
    }
}

__global__ void gmgs_final_kernel(const float* __restrict__ partials, int nparts,
                                  const float* __restrict__ score,
                                  float* __restrict__ out, long long Bdim) {
    __shared__ double s_red[THREADS];
    double acc = 0.0;
    for (int i = threadIdx.x; i < nparts; i += blockDim.x)
        acc += (double)partials[i];
    s_red[threadIdx.x] = acc;
    __syncthreads();
#pragma unroll
    for (int off = THREADS / 2; off > 0; off >>= 1) {
        if (threadIdx.x < off) s_red[threadIdx.x] += s_red[threadIdx.x + off];
        __syncthreads();
    }
    if (threadIdx.x == 0) {
        // common factor: weights were gathered unnormalized; divide once here
        float tot = 0.0f;
#pragma unroll
        for (int i = 0; i < 16; ++i) tot += expf(-score[i]);
        out[0] = (float)(s_red[0] / (double)tot / (double)Bdim);
    }
}

extern "C" void kernel_launch(void* const* d_in, const int* in_sizes, int n_in,
                              void* d_out, int out_size, void* d_ws, size_t ws_size,
                              hipStream_t stream) {
    const float* pred  = (const float*)d_in[0];
    const float* tru   = (const float*)d_in[1];
    const float* score = (const float*)d_in[2];
    float*       out   = (float*)d_out;

    long long n    = (long long)in_sizes[0];   // B*T*1 elements
    long long Bdim = n / 48;                   // T == 48 per reference

    int nb = MAX_BLOCKS;
    size_t wsCap = ws_size / sizeof(float);
    if ((size_t)nb > wsCap) nb = (int)wsCap;
    if (nb < 1) nb = 1;

    float* partials = (float*)d_ws;

    gmgs_partial_kernel<<<nb, THREADS, 0, stream>>>(pred, tru, score, partials, n);
    gmgs_final_kernel<<<1, THREADS, 0, stream>>>(partials, nb, score, out, Bdim);
}